// HistogramNoise_57458072486393
// MI455X (gfx1250) — compile-verified
//
#include <hip/hip_runtime.h>
#include <stdint.h>

typedef __attribute__((ext_vector_type(4))) float floatx4;

#define N_KNOTS 256
#define EVAL_BLOCKS 2048
#define EVAL_THREADS 256

// ---------------------------------------------------------------------------
// Kernel 1: spline moments (not-a-knot, uniform grid) + interval coefficients.
// One block of 256 threads.
//
// Structure trick: subtracting not-a-knot row 0 (M0 - 2M1 + M2 = 0) from
// interior row 1 gives 6*M1 = r1 exactly (same at the far end). The remaining
// unknowns M2..M253 form a 252-eq tridiagonal system (diag 4, off-diag 1),
// solved here by parallel cyclic reduction in double precision: 8 fully
// parallel steps instead of a serial Thomas sweep.
// ---------------------------------------------------------------------------
__global__ __launch_bounds__(256) void spline_setup_kernel(
    const float* __restrict__ y, floatx4* __restrict__ coefs) {
  __shared__ double r[N_KNOTS];
  __shared__ double a[N_KNOTS], b[N_KNOTS], c[N_KNOTS], d[N_KNOTS];
  __shared__ double M[N_KNOTS];

  const int t = threadIdx.x;
  const double h = 1.0 / 255.0;

  // Interior rhs: r_i = 6*(y_{i-1} - 2 y_i + y_{i+1}) / h^2, i = 1..254
  if (t >= 1 && t <= 254) {
    r[t] = 6.0 * ((double)y[t - 1] - 2.0 * (double)y[t] + (double)y[t + 1]) /
           (h * h);
  }
  __syncthreads();

  const double M1 = r[1] / 6.0;      // exact from not-a-knot elimination
  const double M254 = r[254] / 6.0;  // exact from not-a-knot elimination

  // Build equations for unknowns x_t = M[t], t = 2..253:
  //   a x_{t-1} + b x_t + c x_{t+1} = d, with known M1/M254 folded into d.
  // Out-of-range rows are identity equations (x = 0) so PCR padding is exact.
  if (t >= 2 && t <= 253) {
    a[t] = (t == 2) ? 0.0 : 1.0;
    b[t] = 4.0;
    c[t] = (t == 253) ? 0.0 : 1.0;
    double dd = r[t];
    if (t == 2) dd -= M1;
    if (t == 253) dd -= M254;
    d[t] = dd;
  } else {
    a[t] = 0.0;
    b[t] = 1.0;
    c[t] = 0.0;
    d[t] = 0.0;
  }
  __syncthreads();

  // Parallel cyclic reduction: 8 steps (2^8 = 256 >= 252 unknowns).
  for (int delta = 1; delta < N_KNOTS; delta <<= 1) {
    const int im = t - delta, ip = t + delta;
    double am, bm, cm, dm, ap, bp, cp, dp;
    if (im >= 0) {
      am = a[im]; bm = b[im]; cm = c[im]; dm = d[im];
    } else {
      am = 0.0; bm = 1.0; cm = 0.0; dm = 0.0;
    }
    if (ip < N_KNOTS) {
      ap = a[ip]; bp = b[ip]; cp = c[ip]; dp = d[ip];
    } else {
      ap = 0.0; bp = 1.0; cp = 0.0; dp = 0.0;
    }
    const double ai = a[t], bi = b[t], ci = c[t], di = d[t];
    __syncthreads();
    const double alpha = -ai / bm;
    const double gamma = -ci / bp;
    a[t] = alpha * am;
    c[t] = gamma * cp;
    b[t] = bi + alpha * cm + gamma * ap;
    d[t] = di + alpha * dm + gamma * dp;
    __syncthreads();
  }

  // Gather moments.
  double Mt = 0.0;
  if (t >= 2 && t <= 253) Mt = d[t] / b[t];
  if (t == 1) Mt = M1;
  if (t == 254) Mt = M254;
  M[t] = Mt;
  __syncthreads();
  if (t == 0) M[0] = 2.0 * M[1] - M[2];        // not-a-knot boundary
  if (t == 255) M[255] = 2.0 * M[254] - M[253];
  __syncthreads();

  // Per-interval Horner coefficients:
  // f(s) = c0 + s*(c1 + s*(c2 + s*c3)) on interval j, s = t - j*h
  if (t < 255) {
    const double Mj = M[t], Mj1 = M[t + 1];
    const double yj = (double)y[t], yj1 = (double)y[t + 1];
    const double bb = (yj1 - yj) / h - h * (2.0 * Mj + Mj1) / 6.0;
    floatx4 cf;
    cf[0] = (float)yj;
    cf[1] = (float)bb;
    cf[2] = (float)(0.5 * Mj);
    cf[3] = (float)((Mj1 - Mj) / (6.0 * h));
    coefs[t] = cf;
  } else {
    floatx4 cf;
    cf[0] = cf[1] = cf[2] = cf[3] = 0.0f;  // pad entry (j is clamped to 254)
    coefs[t] = cf;
  }
}

// ---------------------------------------------------------------------------
// Kernel 2: streaming evaluation (HBM-bound: 512MB @ 23.3 TB/s ~ 22us).
// The 4KB coefficient table is staged into LDS via the CDNA5 async
// global->LDS DMA path (GLOBAL_LOAD_ASYNC_TO_LDS_B128 / ASYNCcnt).
// ---------------------------------------------------------------------------
__device__ __forceinline__ float eval_one(float x, const floatx4* scoef,
                                          float hf) {
  float t = fminf(fmaxf(x, 0.0f), 1.0f);
  int j = (int)(t * 255.0f);  // trunc == floor (t >= 0)
  j = j > 254 ? 254 : j;
  float s = t - (float)j * hf;
  floatx4 c = scoef[j];
  return c[0] + s * (c[1] + s * (c[2] + s * c[3]));
}

__global__ __launch_bounds__(EVAL_THREADS) void spline_eval_kernel(
    const floatx4* __restrict__ in, floatx4* __restrict__ out,
    const floatx4* __restrict__ coefs, int n4) {
  __shared__ floatx4 scoef[N_KNOTS];

  // Async DMA: each lane moves one 16B table entry straight into LDS.
  {
    uint32_t lds_off = (uint32_t)(uintptr_t)(&scoef[threadIdx.x]);
    uint64_t gaddr = (uint64_t)(uintptr_t)(coefs + threadIdx.x);
    asm volatile("global_load_async_to_lds_b128 %0, %1, off"
                 :
                 : "v"(lds_off), "v"(gaddr)
                 : "memory");
    asm volatile("s_wait_asynccnt 0" ::: "memory");
  }
  __syncthreads();

  const float hf = (float)(1.0 / 255.0);
  const int stride = gridDim.x * blockDim.x;
  int i = blockIdx.x * blockDim.x + threadIdx.x;

  // 2-way unrolled grid-stride loop; NT hints: both streams are touched once
  // and exceed the 192MB L2, so keep them from displacing the hot table.
  for (; i + stride < n4; i += 2 * stride) {
    floatx4 v0 = __builtin_nontemporal_load(&in[i]);
    floatx4 v1 = __builtin_nontemporal_load(&in[i + stride]);
    floatx4 o0, o1;
#pragma unroll
    for (int k = 0; k < 4; ++k) o0[k] = eval_one(v0[k], scoef, hf);
#pragma unroll
    for (int k = 0; k < 4; ++k) o1[k] = eval_one(v1[k], scoef, hf);
    __builtin_nontemporal_store(o0, &out[i]);
    __builtin_nontemporal_store(o1, &out[i + stride]);
  }
  if (i < n4) {
    floatx4 v = __builtin_nontemporal_load(&in[i]);
    floatx4 o;
#pragma unroll
    for (int k = 0; k < 4; ++k) o[k] = eval_one(v[k], scoef, hf);
    __builtin_nontemporal_store(o, &out[i]);
  }
}

// ---------------------------------------------------------------------------
extern "C" void kernel_launch(void* const* d_in, const int* in_sizes, int n_in,
                              void* d_out, int out_size, void* d_ws,
                              size_t ws_size, hipStream_t stream) {
  const float* inp = (const float*)d_in[0];  // "input": 4*256^3 f32
  const float* y = (const float*)d_in[1];    // "y": 256 knot values
  float* out = (float*)d_out;

  floatx4* coefs = (floatx4*)d_ws;  // 256 * 16B = 4KB scratch

  spline_setup_kernel<<<1, 256, 0, stream>>>(y, coefs);

  int n = in_sizes[0];
  int n4 = n / 4;  // 4*256^3 is a multiple of 4
  int blocks = EVAL_BLOCKS;
  int max_blocks = (n4 + EVAL_THREADS - 1) / EVAL_THREADS;
  if (blocks > max_blocks) blocks = max_blocks;

  spline_eval_kernel<<<blocks, EVAL_THREADS, 0, stream>>>(
      (const floatx4*)inp, (floatx4*)out, (const floatx4*)coefs, n4);
}